// MoEGate_71803263255217
// MI455X (gfx1250) — compile-verified
//
#include <hip/hip_runtime.h>
#include <hip/hip_bf16.h>
#include <math.h>
#include <stdint.h>

#define HIDDEN   7168
#define NEXP     256
#define NGRP     8
#define GSIZE    32
#define TOPKG    4
#define TOPK     8
#define KC       64      // K-chunk of A staged in LDS per iteration
#define LSTRIDE  68      // padded float stride: 16B-aligned rows, conflict-free
#define SSTRIDE  257     // padded stride for per-token score rows
#define TOK_TILE 32      // tokens per workgroup (two 16-row M tiles per wave)
#define ABUF     (TOK_TILE * LSTRIDE)   // floats per A double-buffer half

typedef __attribute__((ext_vector_type(2))) float v2f;
typedef __attribute__((ext_vector_type(8))) float v8f;

__launch_bounds__(256)
__global__ void MoEGate_71803263255217_kernel(const float* __restrict__ x,
                                              const float* __restrict__ w,
                                              const float* __restrict__ bias,
                                              int*   __restrict__ out_idx,
                                              float* __restrict__ out_w) {
  __shared__ __align__(16) float A_lds[2 * ABUF];             // ~17.4 KB (double buffered)
  __shared__ __align__(16) float S_lds[TOK_TILE * SSTRIDE];   // ~32.9 KB
  __shared__             float bias_lds[NEXP];

  const int tid  = threadIdx.x;
  const int lane = tid & 31;
  const int wave = tid >> 5;
  const int tok0 = blockIdx.x * TOK_TILE;

  if (tid < NEXP) bias_lds[tid] = bias[tid];

  // Each wave owns experts [32*wave, 32*wave+32) as two 16-wide N tiles.
  const int eb0 = wave * 32;
  const int eb1 = wave * 32 + 16;

  const int row  = lane & 15;   // M index for A frag / N index for B frag
  const int half = lane >> 4;   // selects K pair {0,1} vs {2,3}

  // Per-lane base pointers into the weight rows this lane reads (L2-resident).
  const float* bp0 = &w[(size_t)(eb0 + row) * HIDDEN + 2 * half];
  const float* bp1 = &w[(size_t)(eb1 + row) * HIDDEN + 2 * half];

  // Per-thread A-staging geometry: 512 float4 per chunk, two per thread.
  const int sr0 = tid >> 4;            // token row for i=0   (0..15)
  const int sc0 = tid & 15;            // float4 col
  const int sr1 = 16 + (tid >> 4);     // token row for i=1   (16..31)

  const float* ga0 = &x[(size_t)(tok0 + sr0) * HIDDEN + 4 * sc0];
  const float* ga1 = &x[(size_t)(tok0 + sr1) * HIDDEN + 4 * sc0];
  const unsigned la0 = (unsigned)(uintptr_t)&A_lds[sr0 * LSTRIDE + 4 * sc0];
  const unsigned la1 = (unsigned)(uintptr_t)&A_lds[sr1 * LSTRIDE + 4 * sc0];

  // ---- async-stage chunk 0 into buffer 0 (CDNA5 async-to-LDS, ASYNCcnt)
  {
    const float* g0 = ga0;  const float* g1 = ga1;
    asm volatile("global_load_async_to_lds_b128 %0, %1, off"
                 :: "v"(la0), "v"(g0) : "memory");
    asm volatile("global_load_async_to_lds_b128 %0, %1, off"
                 :: "v"(la1), "v"(g1) : "memory");
  }
  asm volatile("s_wait_asynccnt 0x0" ::: "memory");
  __syncthreads();

  v8f c00 = {}, c01 = {}, c10 = {}, c11 = {};

  for (int kc = 0; kc < HIDDEN; kc += KC) {
    const int cur = (kc / KC) & 1;
    const float* Abuf = &A_lds[cur * ABUF];

    // ---- async-prefetch next A chunk into the other buffer while computing
    if (kc + KC < HIDDEN) {
      const unsigned nb = (cur ^ 1) * ABUF * (unsigned)sizeof(float);
      const unsigned nla0 = la0 + nb;
      const unsigned nla1 = la1 + nb;
      const float* g0 = ga0 + kc + KC;
      const float* g1 = ga1 + kc + KC;
      asm volatile("global_load_async_to_lds_b128 %0, %1, off"
                   :: "v"(nla0), "v"(g0) : "memory");
      asm volatile("global_load_async_to_lds_b128 %0, %1, off"
                   :: "v"(nla1), "v"(g1) : "memory");
    }

    // ---- 16 k-steps; B fragments come straight from global (no LDS round-trip)
    #pragma unroll
    for (int kk = 0; kk < KC; kk += 4) {
      const int kf = kk + 2 * half;
      v2f a0 = *reinterpret_cast<const v2f*>(&Abuf[row * LSTRIDE + kf]);
      v2f a1 = *reinterpret_cast<const v2f*>(&Abuf[(16 + row) * LSTRIDE + kf]);
      v2f b0 = *reinterpret_cast<const v2f*>(&bp0[kc + kk]);
      v2f b1 = *reinterpret_cast<const v2f*>(&bp1[kc + kk]);
      c00 = __builtin_amdgcn_wmma_f32_16x16x4_f32(false, a0, false, b0,
                                                  (short)0, c00, false, false);
      c01 = __builtin_amdgcn_wmma_f32_16x16x4_f32(false, a0, false, b1,
                                                  (short)0, c01, false, false);
      c10 = __builtin_amdgcn_wmma_f32_16x16x4_f32(false, a1, false, b0,
                                                  (short)0, c10, false, false);
      c11 = __builtin_amdgcn_wmma_f32_16x16x4_f32(false, a1, false, b1,
                                                  (short)0, c11, false, false);
    }

    // my async writes to the next buffer are done; barrier publishes them and
    // orders everyone's reads of the current buffer before it is overwritten
    asm volatile("s_wait_asynccnt 0x0" ::: "memory");
    __syncthreads();
  }

  // ---- scatter sigmoid(logits) into per-token score rows in LDS
  // C layout: lane holds N = lane&15; VGPR v holds M = v + 8*(lane>>4)
  {
    const int n = lane & 15;
    #pragma unroll
    for (int v = 0; v < 8; ++v) {
      const int m = v + 8 * half;
      S_lds[m * SSTRIDE + eb0 + n]        = 1.0f / (1.0f + expf(-c00[v]));
      S_lds[m * SSTRIDE + eb1 + n]        = 1.0f / (1.0f + expf(-c01[v]));
      S_lds[(16 + m) * SSTRIDE + eb0 + n] = 1.0f / (1.0f + expf(-c10[v]));
      S_lds[(16 + m) * SSTRIDE + eb1 + n] = 1.0f / (1.0f + expf(-c11[v]));
    }
  }
  __syncthreads();

  // ---- routing: one thread per token (32 of 256 threads)
  if (tid < TOK_TILE) {
    const float* sc = &S_lds[tid * SSTRIDE];

    // group score = sum of top-2 corrected scores per group of 32
    float gs[NGRP];
    #pragma unroll
    for (int g = 0; g < NGRP; ++g) {
      float m1 = -INFINITY, m2 = -INFINITY;
      for (int j = 0; j < GSIZE; ++j) {
        const int e = g * GSIZE + j;
        const float v = sc[e] + bias_lds[e];
        if (v > m1) { m2 = m1; m1 = v; }
        else if (v > m2) { m2 = v; }
      }
      gs[g] = m1 + m2;
    }

    // top-4 groups (strict > -> lower index wins ties, matches jax top_k)
    bool gsel[NGRP];
    #pragma unroll
    for (int g = 0; g < NGRP; ++g) gsel[g] = false;
    for (int r = 0; r < TOPKG; ++r) {
      int best = 0; float bv = -INFINITY;
      for (int g = 0; g < NGRP; ++g)
        if (!gsel[g] && gs[g] > bv) { bv = gs[g]; best = g; }
      gsel[best] = true;
    }

    // top-8 experts over selected groups, ascending index scan + strict
    // insertion keeps jax's stable tie order
    float tv[TOPK]; int ti[TOPK];
    #pragma unroll
    for (int j = 0; j < TOPK; ++j) { tv[j] = -INFINITY; ti[j] = 0; }
    for (int g = 0; g < NGRP; ++g) {
      if (!gsel[g]) continue;
      for (int j = 0; j < GSIZE; ++j) {
        const int e = g * GSIZE + j;
        const float v = sc[e] + bias_lds[e];
        if (v > tv[TOPK - 1]) {
          int p = TOPK - 1;
          while (p > 0 && tv[p - 1] < v) {
            tv[p] = tv[p - 1]; ti[p] = ti[p - 1]; --p;
          }
          tv[p] = v; ti[p] = e;
        }
      }
    }

    // weights from UNcorrected sigmoid scores, normalized * 2.5
    float wsel[TOPK]; float s = 0.0f;
    #pragma unroll
    for (int j = 0; j < TOPK; ++j) { wsel[j] = sc[ti[j]]; s += wsel[j]; }
    const float inv = 2.5f / (s + 1e-20f);

    const size_t base = (size_t)(tok0 + tid) * TOPK;
    #pragma unroll
    for (int j = 0; j < TOPK; ++j) {
      out_idx[base + j] = ti[j];
      out_w[base + j]   = wsel[j] * inv;
    }
  }
}

extern "C" void kernel_launch(void* const* d_in, const int* in_sizes, int n_in,
                              void* d_out, int out_size, void* d_ws, size_t ws_size,
                              hipStream_t stream) {
  const float* x    = (const float*)d_in[0];   // [1, T, 7168] f32
  const float* w    = (const float*)d_in[1];   // [256, 7168] f32
  const float* bias = (const float*)d_in[2];   // [256] f32

  const int T = in_sizes[0] / HIDDEN;          // 8192 tokens

  // Output tuple flattened: topk_idx [T,8] (int32 bits) then topk_weight [T,8] f32
  int*   out_idx = (int*)d_out;
  float* out_w   = (float*)d_out + (size_t)T * TOPK;

  dim3 grid(T / TOK_TILE);
  dim3 block(256);
  MoEGate_71803263255217_kernel<<<grid, block, 0, stream>>>(x, w, bias,
                                                            out_idx, out_w);
}